// TokenChoiceRouting_44117904065240
// MI455X (gfx1250) — compile-verified
//
#include <hip/hip_runtime.h>
#include <hip/hip_bf16.h>

typedef float v2f __attribute__((ext_vector_type(2)));
typedef float v8f __attribute__((ext_vector_type(8)));
typedef unsigned u32x4 __attribute__((ext_vector_type(4)));
typedef int i32x8 __attribute__((ext_vector_type(8)));
typedef int i32x4 __attribute__((ext_vector_type(4)));
typedef int v4i __attribute__((vector_size(16)));   // pointee type for async LDS builtins

#define NUM_EXPERTS 8
#define HDIM        1024
#define NTOK        32768            // B*S = 8*4096
#define CAPACITY    5120             // int(1.25 * 32768 / 8)
#define LBW         0.01f

// workspace layout (32-bit slots)
#define WS_TPE    0   // float[8]  sum of dispatch weights per expert
#define WS_SUMP   8   // float[8]  sum of router probs per expert
#define WS_PREFIX 16  // uint[8]   resolved float-key prefix (radix select)
#define WS_K      24  // uint[8]   remaining count needed within prefix
#define WS_TIE    32  // uint[8]   tie counter for == threshold
#define WS_OVER   40  // uint[8]   expert over capacity?

#if __has_builtin(__builtin_amdgcn_tensor_load_to_lds)
#define HAS_TDM 1
#else
#define HAS_TDM 0
#endif
#if __has_builtin(__builtin_amdgcn_global_load_async_to_lds_b128)
#define HAS_ASYNC 1
#else
#define HAS_ASYNC 0
#endif

typedef __attribute__((address_space(1))) v4i* as1_v4i_p;   // global v4i*
typedef __attribute__((address_space(3))) v4i* as3_v4i_p;   // LDS v4i*
#define AS3P(p) ((__attribute__((address_space(3))) void*)(p))

__device__ __forceinline__ unsigned lds_byte_off(void* p) {
    return (unsigned)(unsigned long long)AS3P(p);
}

__device__ __forceinline__ void wait_asynccnt0() {
#if __has_builtin(__builtin_amdgcn_s_wait_asynccnt)
    __builtin_amdgcn_s_wait_asynccnt(0);
#else
    asm volatile("s_wait_asynccnt 0x0" ::: "memory");
#endif
}

__global__ void zero_stats(float* ws) {
    int t = threadIdx.x;
    if (t < 16) ws[t] = 0.f;       // TPE + SUMP
}

// ---------------------------------------------------------------------------
// Pass 1: router GEMM (WMMA f32 16x16x4) + softmax + top-2 + stats.
// Block = 256 threads = 8 waves; each wave owns a 16-token tile.
// A tiles staged by the Tensor Data Mover (TENSORcnt), router weights by
// async global->LDS copies (ASYNCcnt); sync b128 fallback if builtins absent.
// ---------------------------------------------------------------------------
__global__ __launch_bounds__(256) void router_pass1(
    const float* __restrict__ hidden,   // [NTOK, HDIM]
    const float* __restrict__ rw,       // [8, HDIM]
    float* __restrict__ dispatch,       // [NTOK, 8]
    float* __restrict__ probs_out,      // [NTOK, 8]
    float* __restrict__ ws)
{
    __shared__ float w_lds[16 * 68];        // B tile: 16 expert rows (8 real) x 64 k, pitch 68
    __shared__ float a_lds[8][16 * 68];     // per-wave A tile: 16 tokens x 64 k, pitch 68
    __shared__ float c_lds[8][16 * 9];      // per-wave C scatter: 16 tokens x 8 experts, pitch 9
    __shared__ float s_tpe[NUM_EXPERTS];
    __shared__ float s_prob[NUM_EXPERTS];

    const int tid  = threadIdx.x;
    const int wave = tid >> 5;
    const int lane = tid & 31;
    const int tok_wave = blockIdx.x * 128 + wave * 16;

    // zero fake expert rows 8..15 of B once; zero block stats
    for (int i = tid; i < 16 * 68; i += 256) w_lds[i] = 0.f;
    if (tid < NUM_EXPERTS) { s_tpe[tid] = 0.f; s_prob[tid] = 0.f; }

    const int t16 = lane & 15;   // token row (A) / expert col (B) / N col (C)
    const int kh  = lane >> 4;   // K half per ISA f32 16x4 layout

#if HAS_TDM
    // Wave-uniform (SGPR) pieces of the TDM descriptor.
    const unsigned a_lds_base = __builtin_amdgcn_readfirstlane(
        (int)lds_byte_off(&a_lds[wave][0]));
#endif

    v8f acc = {};

    for (int c = 0; c < 16; ++c) {          // 16 chunks of K=64
        const int k0 = c * 64;
        __syncthreads();                     // protect LDS reuse across chunks

        // ---- stage router weights chunk (rows 0..7; rows 8..15 stay zero)
        if (tid < 128) {
            const int e    = tid >> 4;
            const int koff = (tid & 15) * 4;
#if HAS_ASYNC
            __builtin_amdgcn_global_load_async_to_lds_b128(
                (as1_v4i_p)&rw[e * HDIM + k0 + koff],
                (as3_v4i_p)&w_lds[e * 68 + koff], 0, 0);
#else
            const float4 wv = *(const float4*)&rw[e * HDIM + k0 + koff];
            *(float4*)&w_lds[e * 68 + koff] = wv;
#endif
        }

        // ---- stage this wave's 16x64 A tile
#if HAS_TDM
        {
            // Tensor DMA: 2D tile 64 elems x 16 rows of f32, row stride HDIM,
            // LDS pitch = 64 + 4 dwords via pad (interval 8B<<5=256B, amount 3 -> 4 dw).
            const unsigned long long ga =
                (unsigned long long)&hidden[(size_t)tok_wave * HDIM + k0];
            u32x4 g0;
            g0[0] = 1u;                                        // count=1 valid D#
            g0[1] = a_lds_base;                                // lds_addr
            g0[2] = __builtin_amdgcn_readfirstlane((int)(unsigned)(ga & 0xFFFFFFFFull));
            g0[3] = __builtin_amdgcn_readfirstlane(
                        (int)(((unsigned)(ga >> 32) & 0x01FFFFFFu) | (2u << 30))); // type=2
            i32x8 g1;
            g1[0] = (int)((2u << 16)        // data_size = 4B
                        | (1u << 20)        // pad_enable
                        | (5u << 22)        // pad_interval -> 256B
                        | (3u << 25));      // pad_amount   -> 4 dwords
            g1[1] = (int)(((unsigned)HDIM & 0xFFFFu) << 16);   // tensor_dim0 lo16
            g1[2] = (int)((((unsigned)HDIM >> 16) & 0xFFFFu)
                        | (((unsigned)NTOK & 0xFFFFu) << 16)); // dim0 hi | dim1 lo
            g1[3] = (int)(((unsigned)NTOK >> 16) & 0xFFFFu
                        | (64u << 16));                        // dim1 hi | tile_dim0=64
            g1[4] = 16;                                        // tile_dim1=16, tile_dim2=0
            g1[5] = HDIM;                                      // tensor_dim0_stride lo32
            g1[6] = 0;
            g1[7] = 0;
            const i32x4 gz4 = {0, 0, 0, 0};
            const i32x8 gz8 = {0, 0, 0, 0, 0, 0, 0, 0};
            __builtin_amdgcn_tensor_load_to_lds(g0, g1, gz4, gz4, gz8, 0);
        }
#else
        #pragma unroll
        for (int i = 0; i < 8; ++i) {
            const int f4   = lane + 32 * i;     // 0..255 float4 slots
            const int trow = f4 >> 4;           // token row 0..15
            const int koff = (f4 & 15) * 4;
            const float4 av =
                *(const float4*)&hidden[(size_t)(tok_wave + trow) * HDIM + k0 + koff];
            *(float4*)&a_lds[wave][trow * 68 + koff] = av;
        }
#endif

        // producer waits on its own async/tensor counters, then block barrier
#if HAS_TDM
        __builtin_amdgcn_s_wait_tensorcnt(0);
#endif
#if HAS_ASYNC
        wait_asynccnt0();
#endif
        __syncthreads();

        // ---- 16 WMMA steps of K=4 each (bank-conflict-free b64 LDS reads)
        #pragma unroll
        for (int kk = 0; kk < 16; ++kk) {
            const int off = kk * 4 + 2 * kh;
            v2f a2 = *(const v2f*)&a_lds[wave][t16 * 68 + off];
            v2f b2 = *(const v2f*)&w_lds[t16 * 68 + off];
            acc = __builtin_amdgcn_wmma_f32_16x16x4_f32(
                false, a2, false, b2, (short)0, acc, false, false);
        }
    }

    // scatter C fragment: lane holds col n=t16 of tokens (lane<16 ? 0..7 : 8..15)
    if (t16 < NUM_EXPERTS) {
        const int tbase = (lane < 16) ? 0 : 8;
        #pragma unroll
        for (int v = 0; v < 8; ++v)
            c_lds[wave][(tbase + v) * 9 + t16] = acc[v];
    }
    __syncthreads();

    if (lane < 16) {                         // one lane per token
        const int token = tok_wave + lane;
        float l[8], p[8];
        float mx = -3.4e38f;
        #pragma unroll
        for (int e = 0; e < 8; ++e) { l[e] = c_lds[wave][lane * 9 + e]; mx = fmaxf(mx, l[e]); }
        float s = 0.f;
        #pragma unroll
        for (int e = 0; e < 8; ++e) { p[e] = __expf(l[e] - mx); s += p[e]; }
        const float inv = 1.f / s;
        #pragma unroll
        for (int e = 0; e < 8; ++e) {
            p[e] *= inv;
            probs_out[(size_t)token * 8 + e] = p[e];
            atomicAdd(&s_prob[e], p[e]);
        }
        // top-2 (ties -> lowest index, matching top_k)
        int i1 = 0; float p1 = p[0];
        #pragma unroll
        for (int e = 1; e < 8; ++e) if (p[e] > p1) { p1 = p[e]; i1 = e; }
        int i2 = -1; float p2 = -1.f;
        #pragma unroll
        for (int e = 0; e < 8; ++e) if (e != i1 && p[e] > p2) { p2 = p[e]; i2 = e; }
        const float rs = 1.f / (p1 + p2);
        const float w1 = p1 * rs, w2 = p2 * rs;
        float d[8];
        #pragma unroll
        for (int e = 0; e < 8; ++e) d[e] = (e == i1) ? w1 : ((e == i2) ? w2 : 0.f);
        *(float4*)&dispatch[(size_t)token * 8]     = make_float4(d[0], d[1], d[2], d[3]);
        *(float4*)&dispatch[(size_t)token * 8 + 4] = make_float4(d[4], d[5], d[6], d[7]);
        atomicAdd(&s_tpe[i1], w1);
        atomicAdd(&s_tpe[i2], w2);
    }
    __syncthreads();
    if (tid < NUM_EXPERTS) {
        atomicAdd(&ws[WS_TPE  + tid], s_tpe[tid]);
        atomicAdd(&ws[WS_SUMP + tid], s_prob[tid]);
    }
}

// ---------------------------------------------------------------------------
// LB loss + init radix-select state
// ---------------------------------------------------------------------------
__global__ void lb_setup(const float* wsf, unsigned* wsu, float* lb_out) {
    const int t = threadIdx.x;
    if (t == 0) {
        float acc = 0.f;
        for (int e = 0; e < 8; ++e) acc += wsf[WS_TPE + e] * wsf[WS_SUMP + e];
        // N * sum((tpe/N)*(sump/N)) * w = sum(tpe*sump)/N * w
        lb_out[0] = acc * (LBW / (float)NTOK);
    }
    if (t < NUM_EXPERTS) {
        wsu[WS_OVER   + t] = (wsf[WS_TPE + t] > (float)CAPACITY) ? 1u : 0u;
        wsu[WS_PREFIX + t] = 0u;
        wsu[WS_K      + t] = CAPACITY;
        wsu[WS_TIE    + t] = 0u;
    }
}

// ---------------------------------------------------------------------------
// Radix-select level (12 / 12 / 8 bits of the positive-float key).
// One block per expert; LDS histogram.
// ---------------------------------------------------------------------------
__global__ __launch_bounds__(256) void select_level(
    const float* __restrict__ dispatch, unsigned* wsu, int level)
{
    __shared__ unsigned hist[4096];
    const int e = blockIdx.x;
    if (!wsu[WS_OVER + e]) return;           // uniform per block

    const int      shift = (level == 0) ? 20 : (level == 1) ? 8 : 0;
    const int      bins  = (level == 2) ? 256 : 4096;
    const unsigned pmask = (level == 0) ? 0u
                         : (level == 1) ? 0xFFF00000u : 0xFFFFFF00u;
    const unsigned pval  = wsu[WS_PREFIX + e];

    for (int i = threadIdx.x; i < bins; i += 256) hist[i] = 0;
    __syncthreads();
    for (int n = threadIdx.x; n < NTOK; n += 256) {
        const float w = dispatch[(size_t)n * 8 + e];
        if (w > 0.f) {
            const unsigned key = __float_as_uint(w);
            if ((key & pmask) == pval)
                atomicAdd(&hist[(key >> shift) & (bins - 1)], 1u);
        }
    }
    __syncthreads();
    if (threadIdx.x == 0) {
        unsigned k = wsu[WS_K + e], cum = 0;
        int b;
        for (b = bins - 1; b > 0; --b) {
            if (cum + hist[b] >= k) break;
            cum += hist[b];
        }
        wsu[WS_PREFIX + e] = pval | ((unsigned)b << shift);
        wsu[WS_K + e]      = k - cum;         // still needed within chosen bin
    }
}

// ---------------------------------------------------------------------------
// Apply capacity threshold in place (only over-capacity experts).
// ---------------------------------------------------------------------------
__global__ void apply_capacity(float* dispatch, unsigned* wsu) {
    const int idx = blockIdx.x * blockDim.x + threadIdx.x;
    if (idx >= NTOK * 8) return;
    const int e = idx & 7;
    if (!wsu[WS_OVER + e]) return;
    const float w = dispatch[idx];
    if (w <= 0.f) return;
    const unsigned key = __float_as_uint(w);
    const unsigned T   = wsu[WS_PREFIX + e];
    if (key > T) return;
    if (key < T) { dispatch[idx] = 0.f; return; }
    const unsigned pos = atomicAdd(&wsu[WS_TIE + e], 1u);
    if (pos >= wsu[WS_K + e]) dispatch[idx] = 0.f;
}

// ---------------------------------------------------------------------------
extern "C" void kernel_launch(void* const* d_in, const int* in_sizes, int n_in,
                              void* d_out, int out_size, void* d_ws, size_t ws_size,
                              hipStream_t stream) {
    const float* hidden = (const float*)d_in[0];   // [8,4096,1024] f32
    const float* rw     = (const float*)d_in[1];   // [8,1024] f32

    float* dispatch = (float*)d_out;                     // [N,8]
    float* lb       = dispatch + (size_t)NTOK * 8;       // scalar
    float* probs    = lb + 1;                            // [N,8]

    float*    wsf = (float*)d_ws;
    unsigned* wsu = (unsigned*)d_ws;

    zero_stats<<<1, 32, 0, stream>>>(wsf);
    router_pass1<<<NTOK / 128, 256, 0, stream>>>(hidden, rw, dispatch, probs, wsf);
    lb_setup<<<1, 32, 0, stream>>>(wsf, wsu, lb);
    for (int lvl = 0; lvl < 3; ++lvl)
        select_level<<<NUM_EXPERTS, 256, 0, stream>>>(dispatch, wsu, lvl);
    apply_capacity<<<(NTOK * 8 + 255) / 256, 256, 0, stream>>>(dispatch, wsu);
}